// MPNNMolNet_84000970375665
// MI455X (gfx1250) — compile-verified
//
#include <hip/hip_runtime.h>
#include <hip/hip_bf16.h>

// ---------------- problem constants (match reference) ----------------
#define IN_DIM   32
#define EDGE_DIM 8
#define HID      64
#define N_TASKS  12
#define NNODES   50000
#define NEDGES   100000
#define NGRAPHS  2000
#define BN_EPS   1e-5f

typedef __attribute__((ext_vector_type(2))) float v2f;
typedef __attribute__((ext_vector_type(8))) float v8f;
typedef int v4i __attribute__((vector_size(16)));   // matches builtin param type

// ---------------------------------------------------------------------
// gfx1250 async global->LDS copy (ASYNCcnt-tracked, no VGPR round-trip).
// ROCm declares the builtin as (v4i AS1*, v4i AS3*, imm, imm).  Kernel-arg
// pointers are global addresses; generic LDS pointers carry the LDS byte
// offset in their low 32 bits, so the integer casts below produce the
// AS1/AS3 pointers the builtin wants.
// ---------------------------------------------------------------------
__device__ __forceinline__ void async_copy_b128(const void* gsrc, void* ldst)
{
#if __has_builtin(__builtin_amdgcn_global_load_async_to_lds_b128)
    __builtin_amdgcn_global_load_async_to_lds_b128(
        (__attribute__((address_space(1))) v4i*)(uintptr_t)gsrc,
        (__attribute__((address_space(3))) v4i*)(uint32_t)(uintptr_t)ldst,
        0, 0);
#else
    *(float4*)ldst = *(const float4*)gsrc;
#endif
}

__device__ __forceinline__ void async_wait_all()
{
#if __has_builtin(__builtin_amdgcn_s_wait_asynccnt)
    __builtin_amdgcn_s_wait_asynccnt(0);
#else
    asm volatile("s_wait_asynccnt 0" ::: "memory");
#endif
}

// ---------------------------------------------------------------------
// Fused NNConv edge kernel.
//
// One wave32 owns a PAIR of 16-edge tiles (32 edges) so the A operand
// (We slice) and the C init (be) -- identical for every edge tile -- are
// loaded once per (i, ob) step and feed FOUR V_WMMA_F32_16X16X4_F32 ops
// (2 tiles x 2 K-halves).  The transposed theta tile
//    D[o, e] = relu( sum_k We[k][i*64 + ob*16 + o] * ea[e][k] + be[...] )
// is produced in registers; lane l holds edge e = l%16 and rows
// o' = v + 8*(l>>4), so the per-edge scale x[src[e]][i] is one per-lane
// scalar and  m[e][o] += xs * relu(D)  needs no cross-lane traffic.
//
// We+be are staged in LDS per workgroup via async-to-LDS (in_ch=64:
// 128KB+16KB of the 320KB WGP LDS).  We rows are padded by +16 floats so
// the two lane halves (k-row offset 2*stride) hit disjoint bank halves.
//
// Finally 32 global fp32 atomics per lane per tile scatter m into
// aggr[dst].  E must be a multiple of 32 (100000 is).
// ---------------------------------------------------------------------
template <int IN_CH>
__global__ __launch_bounds__(512)
void edge_nnconv_wmma(const float* __restrict__ h_in,   // [N, IN_CH]
                      const int*   __restrict__ ei,     // [2, E]
                      const float* __restrict__ ea,     // [E, 8]
                      const float* __restrict__ We,     // [8, IN_CH*64]
                      const float* __restrict__ be,     // [IN_CH*64]
                      float*       __restrict__ aggr,   // [N, 64]
                      int E)
{
    constexpr int S  = IN_CH * HID;      // We row length (floats)
    constexpr int SP = S + 16;           // padded LDS row stride (bank skew)
    __shared__ __align__(16) float sWe[8 * SP];
    __shared__ __align__(16) float sbe[S];

    // ---- async-stage We (padded rows) + be into LDS, 16B granules ----
    {
        const float4* We4 = (const float4*)We;
        for (int idx = threadIdx.x; idx < 8 * (S / 4); idx += blockDim.x) {
            const int k = idx / (S / 4);
            const int j = idx - k * (S / 4);
            async_copy_b128(&We4[idx], &sWe[k * SP + 4 * j]);
        }
        const float4* be4 = (const float4*)be;
        for (int idx = threadIdx.x; idx < S / 4; idx += blockDim.x)
            async_copy_b128(&be4[idx], &sbe[4 * idx]);
    }
    async_wait_all();
    __syncthreads();

    const int lane   = threadIdx.x & 31;
    const int wave   = threadIdx.x >> 5;
    const int col16  = lane & 15;            // edge slot / A row M
    const int hi     = lane >> 4;            // lane half
    const int nWaves = gridDim.x * (blockDim.x >> 5);
    const int nPairs = E / 32;

    for (int p = blockIdx.x * (blockDim.x >> 5) + wave; p < nPairs; p += nWaves) {
        const int e0 = p * 32 + col16;       // tile 0 edge
        const int e1 = e0 + 16;              // tile 1 edge

        // prefetch next pair's edge attributes (global_prefetch_b8)
        if (p + nWaves < nPairs)
            __builtin_prefetch(ea + (size_t)(p + nWaves) * 32 * EDGE_DIM, 0, 1);

        const int src0 = ei[e0], dst0 = ei[E + e0];
        const int src1 = ei[e1], dst1 = ei[E + e1];

        // B operands (4x16 K x edges): VGPR v, lane -> K = v + 2*hi, N = l%16
        const float2 b00 = *(const float2*)(ea + (size_t)e0 * EDGE_DIM + 2 * hi);
        const float2 b01 = *(const float2*)(ea + (size_t)e0 * EDGE_DIM + 4 + 2 * hi);
        const float2 b10 = *(const float2*)(ea + (size_t)e1 * EDGE_DIM + 2 * hi);
        const float2 b11 = *(const float2*)(ea + (size_t)e1 * EDGE_DIM + 4 + 2 * hi);
        v2f B00; B00[0] = b00.x; B00[1] = b00.y;   // tile0, K = 0..3
        v2f B01; B01[0] = b01.x; B01[1] = b01.y;   // tile0, K = 4..7
        v2f B10; B10[0] = b10.x; B10[1] = b10.y;   // tile1, K = 0..3
        v2f B11; B11[0] = b11.x; B11[1] = b11.y;   // tile1, K = 4..7

        v8f acc0[4], acc1[4];
        #pragma unroll
        for (int ob = 0; ob < 4; ++ob) { acc0[ob] = (v8f)(0.0f); acc1[ob] = (v8f)(0.0f); }

        const float* xrow0 = h_in + (size_t)src0 * IN_CH;
        const float* xrow1 = h_in + (size_t)src1 * IN_CH;

        #pragma unroll 4
        for (int i = 0; i < IN_CH; ++i) {
            const float xs0     = xrow0[i];
            const float xs1     = xrow1[i];
            const int   colBase = i * HID;
            #pragma unroll
            for (int ob = 0; ob < 4; ++ob) {
                const int col = colBase + ob * 16 + col16;
                // A operand from LDS: lane -> M = l%16, K = v + 2*hi
                v2f A0, A1;
                A0[0] = sWe[(2 * hi + 0) * SP + col];
                A0[1] = sWe[(2 * hi + 1) * SP + col];
                A1[0] = sWe[(2 * hi + 4) * SP + col];
                A1[1] = sWe[(2 * hi + 5) * SP + col];
                // C init = be broadcast across edge columns (rows M = v+8*hi)
                const float* bp = sbe + colBase + ob * 16 + 8 * hi;
                const float4 c0q = *(const float4*)(bp);
                const float4 c1q = *(const float4*)(bp + 4);
                const v8f cini = { c0q.x, c0q.y, c0q.z, c0q.w,
                                   c1q.x, c1q.y, c1q.z, c1q.w };
                // two independent WMMA chains (tile0 / tile1), K=8 as 2x K=4
                v8f c0 = __builtin_amdgcn_wmma_f32_16x16x4_f32(false, A0, false, B00,
                                                               (short)0, cini, false, false);
                v8f c1 = __builtin_amdgcn_wmma_f32_16x16x4_f32(false, A0, false, B10,
                                                               (short)0, cini, false, false);
                c0 = __builtin_amdgcn_wmma_f32_16x16x4_f32(false, A1, false, B01,
                                                           (short)0, c0, false, false);
                c1 = __builtin_amdgcn_wmma_f32_16x16x4_f32(false, A1, false, B11,
                                                           (short)0, c1, false, false);
                // m[e][o] += xs * relu(theta)
                #pragma unroll
                for (int v = 0; v < 8; ++v) {
                    acc0[ob][v] = fmaf(xs0, fmaxf(c0[v], 0.0f), acc0[ob][v]);
                    acc1[ob][v] = fmaf(xs1, fmaxf(c1[v], 0.0f), acc1[ob][v]);
                }
            }
        }

        // scatter-add messages; lane l covers o = ob*16 + v + 8*hi
        float* arow0 = aggr + (size_t)dst0 * HID;
        float* arow1 = aggr + (size_t)dst1 * HID;
        #pragma unroll
        for (int ob = 0; ob < 4; ++ob)
            #pragma unroll
            for (int v = 0; v < 8; ++v) {
                atomicAdd(&arow0[ob * 16 + v + 8 * hi], acc0[ob][v]);
                atomicAdd(&arow1[ob * 16 + v + 8 * hi], acc1[ob][v]);
            }
    }
}

// ---------------------------------------------------------------------
__global__ void zero_f32(float* __restrict__ p, int n)
{
    int i = blockIdx.x * blockDim.x + threadIdx.x;
    const int stride = gridDim.x * blockDim.x;
    for (; i < n; i += stride) p[i] = 0.0f;
}

__global__ void count_edges(const int* __restrict__ ei, float* __restrict__ cnt, int E)
{
    int e = blockIdx.x * blockDim.x + threadIdx.x;
    if (e < E) atomicAdd(&cnt[ei[E + e]], 1.0f);
}

__global__ void count_graphs(const int* __restrict__ batch, float* __restrict__ gcnt, int N)
{
    int v = blockIdx.x * blockDim.x + threadIdx.x;
    if (v < N) atomicAdd(&gcnt[batch[v]], 1.0f);
}

// h_pre[v][o] = aggr[v][o]/max(cnt,1) + x[v]@root[:,o] + bias[o]   (in place OK)
template <int IN_CH>
__global__ void node_update(const float* __restrict__ aggr,
                            const float* __restrict__ h_in,
                            const float* __restrict__ root,   // [IN_CH, 64]
                            const float* __restrict__ bias,   // [64]
                            const float* __restrict__ cnt,
                            float*       __restrict__ h_out,  // may alias aggr
                            int N)
{
    int idx = blockIdx.x * blockDim.x + threadIdx.x;
    if (idx >= N * HID) return;
    const int v = idx >> 6;
    const int o = idx & 63;
    float s = aggr[idx] / fmaxf(cnt[v], 1.0f);
    const float* xr = h_in + (size_t)v * IN_CH;
    float a = bias[o];
    #pragma unroll 8
    for (int i = 0; i < IN_CH; ++i)
        a = fmaf(xr[i], root[i * HID + o], a);
    h_out[idx] = s + a;
}

// per-column sums / sums of squares (block-reduced, then few atomics)
__global__ __launch_bounds__(256)
void bn_stats(const float* __restrict__ h, int N,
              float* __restrict__ sum, float* __restrict__ sumsq)
{
    __shared__ float ls[4][HID];
    __shared__ float lq[4][HID];
    const int o   = threadIdx.x & 63;
    const int sub = threadIdx.x >> 6;
    float s = 0.0f, q = 0.0f;
    const int rows = gridDim.x * 4;
    for (int v = blockIdx.x * 4 + sub; v < N; v += rows) {
        const float x = h[(size_t)v * HID + o];
        s += x;
        q  = fmaf(x, x, q);
    }
    ls[sub][o] = s;  lq[sub][o] = q;
    __syncthreads();
    if (sub == 0) {
        s = ls[0][o] + ls[1][o] + ls[2][o] + ls[3][o];
        q = lq[0][o] + lq[1][o] + lq[2][o] + lq[3][o];
        atomicAdd(&sum[o], s);
        atomicAdd(&sumsq[o], q);
    }
}

__global__ void bn_apply(float* __restrict__ h,
                         const float* __restrict__ sum,
                         const float* __restrict__ sumsq,
                         const float* __restrict__ gamma,
                         const float* __restrict__ beta, int N)
{
    int idx = blockIdx.x * blockDim.x + threadIdx.x;
    if (idx >= N * HID) return;
    const int o   = idx & 63;
    const float invN = 1.0f / (float)N;
    const float mu  = sum[o] * invN;
    const float var = sumsq[o] * invN - mu * mu;  // biased variance
    const float y = (h[idx] - mu) * rsqrtf(var + BN_EPS) * gamma[o] + beta[o];
    h[idx] = fmaxf(y, 0.0f);
}

__global__ void pool_nodes(const float* __restrict__ h,
                           const int*   __restrict__ batch,
                           float*       __restrict__ pooled, int N)
{
    int idx = blockIdx.x * blockDim.x + threadIdx.x;
    if (idx >= N * HID) return;
    const int v = idx >> 6;
    const int o = idx & 63;
    atomicAdd(&pooled[(size_t)batch[v] * HID + o], h[idx]);
}

__global__ void head_kernel(const float* __restrict__ pooled,
                            const float* __restrict__ gcnt,
                            const float* __restrict__ Wh,   // [64, 12]
                            const float* __restrict__ bh,   // [12]
                            float*       __restrict__ out, int G)
{
    int idx = blockIdx.x * blockDim.x + threadIdx.x;
    if (idx >= G * N_TASKS) return;
    const int g = idx / N_TASKS;
    const int t = idx - g * N_TASKS;
    const float inv = 1.0f / fmaxf(gcnt[g], 1.0f);
    const float* pr = pooled + (size_t)g * HID;
    float a = bh[t];
    #pragma unroll 8
    for (int o = 0; o < HID; ++o)
        a = fmaf(pr[o] * inv, Wh[o * N_TASKS + t], a);
    out[idx] = a;
}

// ---------------------------------------------------------------------
extern "C" void kernel_launch(void* const* d_in, const int* in_sizes, int n_in,
                              void* d_out, int out_size, void* d_ws, size_t ws_size,
                              hipStream_t stream)
{
    (void)in_sizes; (void)n_in; (void)out_size; (void)ws_size;

    const float* x   = (const float*)d_in[0];
    const int*   ei  = (const int*)  d_in[1];
    const float* ea  = (const float*)d_in[2];
    const int*   bat = (const int*)  d_in[3];
    const float* We[3] = {(const float*)d_in[4],  (const float*)d_in[10], (const float*)d_in[16]};
    const float* be[3] = {(const float*)d_in[5],  (const float*)d_in[11], (const float*)d_in[17]};
    const float* rt[3] = {(const float*)d_in[6],  (const float*)d_in[12], (const float*)d_in[18]};
    const float* bi[3] = {(const float*)d_in[7],  (const float*)d_in[13], (const float*)d_in[19]};
    const float* ga[3] = {(const float*)d_in[8],  (const float*)d_in[14], (const float*)d_in[20]};
    const float* bt[3] = {(const float*)d_in[9],  (const float*)d_in[15], (const float*)d_in[21]};
    const float* Wh  = (const float*)d_in[22];
    const float* bh  = (const float*)d_in[23];
    float* out = (float*)d_out;

    // workspace partition (~26.3 MB of f32)
    float* ws     = (float*)d_ws;
    float* hbuf0  = ws;                                  // NNODES*HID
    float* hbuf1  = hbuf0 + (size_t)NNODES * HID;        // NNODES*HID
    float* cnt    = hbuf1 + (size_t)NNODES * HID;        // NNODES
    float* gcnt   = cnt + NNODES;                        // NGRAPHS
    float* pooled = gcnt + NGRAPHS;                      // NGRAPHS*HID
    float* bsum   = pooled + (size_t)NGRAPHS * HID;      // HID
    float* bsq    = bsum + HID;                          // HID

    // one-time degree / graph-size counts + pooled init
    zero_f32<<<256, 256, 0, stream>>>(cnt, NNODES);
    zero_f32<<<64, 256, 0, stream>>>(gcnt, NGRAPHS);
    zero_f32<<<512, 256, 0, stream>>>(pooled, NGRAPHS * HID);
    count_edges<<<(NEDGES + 255) / 256, 256, 0, stream>>>(ei, cnt, NEDGES);
    count_graphs<<<(NNODES + 255) / 256, 256, 0, stream>>>(bat, gcnt, NNODES);

    const int nelems = NNODES * HID;
    const int eblks  = 128;                  // persistent: 128 blocks x 16 waves

    const float* hin = x;
    for (int l = 0; l < 3; ++l) {
        float* aggr = (l == 1) ? hbuf1 : hbuf0;   // ping-pong, never aliases hin
        zero_f32<<<2048, 256, 0, stream>>>(aggr, nelems);
        zero_f32<<<1, 128, 0, stream>>>(bsum, 2 * HID);   // bsum+bsq contiguous

        if (l == 0) {
            edge_nnconv_wmma<IN_DIM><<<eblks, 512, 0, stream>>>(
                hin, ei, ea, We[l], be[l], aggr, NEDGES);
            node_update<IN_DIM><<<(nelems + 255) / 256, 256, 0, stream>>>(
                aggr, hin, rt[l], bi[l], cnt, aggr, NNODES);
        } else {
            edge_nnconv_wmma<HID><<<eblks, 512, 0, stream>>>(
                hin, ei, ea, We[l], be[l], aggr, NEDGES);
            node_update<HID><<<(nelems + 255) / 256, 256, 0, stream>>>(
                aggr, hin, rt[l], bi[l], cnt, aggr, NNODES);
        }
        bn_stats<<<256, 256, 0, stream>>>(aggr, NNODES, bsum, bsq);
        bn_apply<<<(nelems + 255) / 256, 256, 0, stream>>>(
            aggr, bsum, bsq, ga[l], bt[l], NNODES);
        hin = aggr;
    }

    pool_nodes<<<(nelems + 255) / 256, 256, 0, stream>>>(hin, bat, pooled, NNODES);
    head_kernel<<<(NGRAPHS * N_TASKS + 255) / 256, 256, 0, stream>>>(
        pooled, gcnt, Wh, bh, out, NGRAPHS);
}